// SybilGAT_6803228196961
// MI455X (gfx1250) — compile-verified
//
#include <hip/hip_runtime.h>

typedef __attribute__((ext_vector_type(2))) float v2f;
typedef __attribute__((ext_vector_type(8))) float v8f;

namespace {

constexpr float kSlope = 0.2f;  // leaky_relu negative slope

// Order-preserving float <-> uint mapping so segment-max can use atomicMax(u32).
__device__ __forceinline__ unsigned fenc(float x) {
  unsigned u = __float_as_uint(x);
  return (u & 0x80000000u) ? ~u : (u | 0x80000000u);
}
__device__ __forceinline__ float fdec(unsigned u) {
  unsigned v = (u & 0x80000000u) ? (u & 0x7FFFFFFFu) : ~u;
  return __uint_as_float(v);
}
__device__ __forceinline__ float lrelu(float x) { return x > 0.f ? x : kSlope * x; }

// ---------------------------------------------------------------------------
// Zero-fill for the atomic-accumulator region (must run every call).
__global__ void zero_f(float* __restrict__ p, long long n) {
  long long i = (long long)blockIdx.x * blockDim.x + threadIdx.x;
  long long st = (long long)gridDim.x * blockDim.x;
  for (; i < n; i += st) p[i] = 0.f;
}

// ---------------------------------------------------------------------------
// GEMM: Y[N,NOUT] = X[N,K] @ W[K,NOUT], f32 via V_WMMA_F32_16X16X4_F32.
// One wave per 16x16 tile. Block = (NOUT/16) waves, blockIdx.x = M tile.
// A 16x4 per-lane layout: lanes 0-15 hold K=k,k+1 ; lanes 16-31 hold K=k+2,k+3.
// B 4x16: v0/v1 hold rows K=k..k+1 (lanes 0-15) and K=k+2..k+3 (lanes 16-31).
// C/D 16x16: VGPR g = row m0+g (lanes 0-15) / m0+8+g (lanes 16-31), col = n0+l16.
template <int K, int NOUT>
__global__ void gemm_wmma_f32(const float* __restrict__ X, const float* __restrict__ W,
                              float* __restrict__ Y, int N) {
  const int m0   = blockIdx.x * 16;
  const int n0   = (threadIdx.x >> 5) * 16;
  const int lane = threadIdx.x & 31;
  const int half = lane >> 4;
  const int l16  = lane & 15;

  int rowA = m0 + l16;
  if (rowA >= N) rowA = N - 1;  // clamp (harmless duplicate load)

  const float* __restrict__ arow = X + (size_t)rowA * K + 2 * half;
  const float* __restrict__ bcol = W + (size_t)(2 * half) * NOUT + n0 + l16;

  v8f c = {};
#pragma unroll
  for (int k = 0; k < K; k += 4) {
    v2f a;
    a.x = arow[k];
    a.y = arow[k + 1];
    v2f b;
    b.x = bcol[(size_t)k * NOUT];
    b.y = bcol[(size_t)(k + 1) * NOUT];
    c = __builtin_amdgcn_wmma_f32_16x16x4_f32(false, a, false, b, (short)0, c, false, false);
  }

  const int rbase = m0 + 8 * half;
  float* __restrict__ yp = Y + (size_t)rbase * NOUT + n0 + l16;
#pragma unroll
  for (int g = 0; g < 8; ++g) {
    if (rbase + g < N) yp[(size_t)g * NOUT] = c[g];
  }
}

// ---------------------------------------------------------------------------
// Per-(node, head) attention scalars: a_src[n,h] = <h[n,h,:], att_src[h,:]>.
template <int H, int C>
__global__ void att_scores(const float* __restrict__ Hm, const float* __restrict__ as,
                           const float* __restrict__ ad, float* __restrict__ asr,
                           float* __restrict__ adt, int N) {
  int t = blockIdx.x * blockDim.x + threadIdx.x;
  if (t >= N * H) return;
  const int n = t / H, hh = t % H;
  const float* __restrict__ hp = Hm + (size_t)n * (H * C) + hh * C;
  float s = 0.f, d = 0.f;
#pragma unroll
  for (int c = 0; c < C; ++c) {
    float v = hp[c];
    s += v * as[hh * C + c];
    d += v * ad[hh * C + c];
  }
  asr[t] = s;
  adt[t] = d;
}

// ---------------------------------------------------------------------------
// Edge pass 1: segment max over dst (encoded-float atomicMax).
template <int H>
__global__ void edge_max(const int* __restrict__ ei, int E, int EN,
                         const float* __restrict__ asr, const float* __restrict__ adt,
                         unsigned* __restrict__ mx) {
  long long t = (long long)blockIdx.x * blockDim.x + threadIdx.x;
  if (t >= (long long)EN * H) return;
  const int e = (int)(t / H), hh = (int)(t % H);
  int s, d;
  if (e < E) { s = ei[e]; d = ei[E + e]; } else { s = d = e - E; }
  float sc = lrelu(asr[s * H + hh] + adt[d * H + hh]);
  atomicMax(&mx[d * H + hh], fenc(sc));
}

// Edge pass 2: sum of exp(score - max[dst]).
template <int H>
__global__ void edge_expsum(const int* __restrict__ ei, int E, int EN,
                            const float* __restrict__ asr, const float* __restrict__ adt,
                            const unsigned* __restrict__ mx, float* __restrict__ sm) {
  long long t = (long long)blockIdx.x * blockDim.x + threadIdx.x;
  if (t >= (long long)EN * H) return;
  const int e = (int)(t / H), hh = (int)(t % H);
  int s, d;
  if (e < E) { s = ei[e]; d = ei[E + e]; } else { s = d = e - E; }
  float sc = lrelu(asr[s * H + hh] + adt[d * H + hh]);
  atomicAdd(&sm[d * H + hh], __expf(sc - fdec(mx[d * H + hh])));
}

// Edge pass 3: agg[dst] += alpha * h[src]. One wave per edge, VPL channels/lane.
template <int H, int C, int VPL>
__global__ void edge_aggr(const int* __restrict__ ei, int E, int EN,
                          const float* __restrict__ Hm, const float* __restrict__ asr,
                          const float* __restrict__ adt, const unsigned* __restrict__ mx,
                          const float* __restrict__ sm, float* __restrict__ agg) {
  constexpr int HC = H * C;
  long long t = (long long)blockIdx.x * blockDim.x + threadIdx.x;
  const int lane = (int)(t & 31);
  const long long e = t >> 5;
  if (e >= EN) return;
  int s, d;
  if (e < E) { s = ei[e]; d = ei[E + e]; } else { s = d = (int)(e - E); }
  const int ch0 = lane * VPL;
  const int hh = ch0 / C;
  float sc = lrelu(asr[s * H + hh] + adt[d * H + hh]);
  float alpha = __expf(sc - fdec(mx[d * H + hh])) / sm[d * H + hh];
  const float* __restrict__ hp = Hm + (size_t)s * HC + ch0;
  float* __restrict__ op = agg + (size_t)d * HC + ch0;
#pragma unroll
  for (int k = 0; k < VPL; ++k) atomicAdd(op + k, hp[k] * alpha);
}

// y = elu(y + bias[i & mask]) in place (HC is a power of two).
__global__ void bias_elu(float* __restrict__ y, const float* __restrict__ b, int total,
                         int mask) {
  int t = blockIdx.x * blockDim.x + threadIdx.x;
  if (t >= total) return;
  float v = y[t] + b[t & mask];
  y[t] = v > 0.f ? v : (__expf(v) - 1.f);
}

// ---------------------------------------------------------------------------
// Layer 3 node kernel: g3 = x3 @ W3 (64->2) plus attention scalars (H=1,C=2).
__global__ void l3_node(const float* __restrict__ x3, const float* __restrict__ W3,
                        const float* __restrict__ as3, const float* __restrict__ ad3,
                        float* __restrict__ g3, float* __restrict__ asr,
                        float* __restrict__ adt, int N) {
  int n = blockIdx.x * blockDim.x + threadIdx.x;
  if (n >= N) return;
  const float* __restrict__ xp = x3 + (size_t)n * 64;
  float o0 = 0.f, o1 = 0.f;
#pragma unroll
  for (int k = 0; k < 64; ++k) {
    float v = xp[k];
    o0 += v * W3[2 * k];
    o1 += v * W3[2 * k + 1];
  }
  g3[2 * n] = o0;
  g3[2 * n + 1] = o1;
  asr[n] = o0 * as3[0] + o1 * as3[1];
  adt[n] = o0 * ad3[0] + o1 * ad3[1];
}

// Layer 3 aggregation: one thread per edge, 2 channels.
__global__ void edge_aggr2ch(const int* __restrict__ ei, int E, int EN,
                             const float* __restrict__ g3, const float* __restrict__ asr,
                             const float* __restrict__ adt, const unsigned* __restrict__ mx,
                             const float* __restrict__ sm, float* __restrict__ agg) {
  int e = blockIdx.x * blockDim.x + threadIdx.x;
  if (e >= EN) return;
  int s, d;
  if (e < E) { s = ei[e]; d = ei[E + e]; } else { s = d = e - E; }
  float sc = lrelu(asr[s] + adt[d]);
  float alpha = __expf(sc - fdec(mx[d])) / sm[d];
  atomicAdd(&agg[2 * d], g3[2 * s] * alpha);
  atomicAdd(&agg[2 * d + 1], g3[2 * s + 1] * alpha);
}

__global__ void final_logsoftmax(const float* __restrict__ agg3, const float* __restrict__ b3,
                                 float* __restrict__ out, int N) {
  int n = blockIdx.x * blockDim.x + threadIdx.x;
  if (n >= N) return;
  float z0 = agg3[2 * n] + b3[0];
  float z1 = agg3[2 * n + 1] + b3[1];
  float m = fmaxf(z0, z1);
  float lse = m + __logf(__expf(z0 - m) + __expf(z1 - m));
  out[2 * n] = z0 - lse;
  out[2 * n + 1] = z1 - lse;
}

inline int cdiv(long long a, long long b) { return (int)((a + b - 1) / b); }

}  // namespace

extern "C" void kernel_launch(void* const* d_in, const int* in_sizes, int n_in,
                              void* d_out, int out_size, void* d_ws, size_t ws_size,
                              hipStream_t stream) {
  const float* x   = (const float*)d_in[0];
  const int*   ei  = (const int*)d_in[1];
  const float* W1  = (const float*)d_in[2];
  const float* as1 = (const float*)d_in[3];
  const float* ad1 = (const float*)d_in[4];
  const float* b1  = (const float*)d_in[5];
  const float* W2  = (const float*)d_in[6];
  const float* as2 = (const float*)d_in[7];
  const float* ad2 = (const float*)d_in[8];
  const float* b2  = (const float*)d_in[9];
  const float* W3  = (const float*)d_in[10];
  const float* as3 = (const float*)d_in[11];
  const float* ad3 = (const float*)d_in[12];
  const float* b3  = (const float*)d_in[13];

  const int N  = in_sizes[0] / 128;  // 50000
  const int E  = in_sizes[1] / 2;    // 1600000
  const int EN = E + N;              // edges incl. self-loops

  // Workspace layout (floats). All atomically-updated buffers live in one
  // contiguous region starting at `zoff` and are zeroed each call.
  float* w = (float*)d_ws;
  size_t o = 0;
  float* h1   = w + o; o += (size_t)N * 128;  // layer1 GEMM output
  float* g2   = w + o; o += (size_t)N * 64;   // layer2 GEMM output
  float* g3   = w + o; o += (size_t)N * 2;    // layer3 GEMM output
  float* asr  = w + o; o += (size_t)N * 4;    // a_src (reused per layer)
  float* adt  = w + o; o += (size_t)N * 4;    // a_dst (reused per layer)
  const size_t zoff = o;
  float* agg1 = w + o; o += (size_t)N * 128;
  float* agg2 = w + o; o += (size_t)N * 64;
  float* agg3 = w + o; o += (size_t)N * 2;
  float* mx1  = w + o; o += (size_t)N * 4;
  float* sm1  = w + o; o += (size_t)N * 4;
  float* mx2  = w + o; o += (size_t)N * 2;
  float* sm2  = w + o; o += (size_t)N * 2;
  float* mx3  = w + o; o += (size_t)N * 1;
  float* sm3  = w + o; o += (size_t)N * 1;
  const long long zcount = (long long)(o - zoff);

  // 0) zero accumulators / softmax state (required every call for replay)
  zero_f<<<4096, 256, 0, stream>>>(w + zoff, zcount);

  // ---- Layer 1: 128 -> 4 heads x 32, concat ----
  gemm_wmma_f32<128, 128><<<cdiv(N, 16), 8 * 32, 0, stream>>>(x, W1, h1, N);
  att_scores<4, 32><<<cdiv((long long)N * 4, 256), 256, 0, stream>>>(h1, as1, ad1, asr, adt, N);
  edge_max<4><<<cdiv((long long)EN * 4, 256), 256, 0, stream>>>(ei, E, EN, asr, adt, (unsigned*)mx1);
  edge_expsum<4><<<cdiv((long long)EN * 4, 256), 256, 0, stream>>>(ei, E, EN, asr, adt,
                                                                  (const unsigned*)mx1, sm1);
  edge_aggr<4, 32, 4><<<cdiv((long long)EN * 32, 256), 256, 0, stream>>>(
      ei, E, EN, h1, asr, adt, (const unsigned*)mx1, sm1, agg1);
  bias_elu<<<cdiv((long long)N * 128, 256), 256, 0, stream>>>(agg1, b1, N * 128, 127);

  // ---- Layer 2: 128 -> 2 heads x 32, concat ----
  gemm_wmma_f32<128, 64><<<cdiv(N, 16), 4 * 32, 0, stream>>>(agg1, W2, g2, N);
  att_scores<2, 32><<<cdiv((long long)N * 2, 256), 256, 0, stream>>>(g2, as2, ad2, asr, adt, N);
  edge_max<2><<<cdiv((long long)EN * 2, 256), 256, 0, stream>>>(ei, E, EN, asr, adt, (unsigned*)mx2);
  edge_expsum<2><<<cdiv((long long)EN * 2, 256), 256, 0, stream>>>(ei, E, EN, asr, adt,
                                                                  (const unsigned*)mx2, sm2);
  edge_aggr<2, 32, 2><<<cdiv((long long)EN * 32, 256), 256, 0, stream>>>(
      ei, E, EN, g2, asr, adt, (const unsigned*)mx2, sm2, agg2);
  bias_elu<<<cdiv((long long)N * 64, 256), 256, 0, stream>>>(agg2, b2, N * 64, 63);

  // ---- Layer 3: 64 -> 2, 1 head, mean(=identity) + log_softmax ----
  l3_node<<<cdiv(N, 256), 256, 0, stream>>>(agg2, W3, as3, ad3, g3, asr, adt, N);
  edge_max<1><<<cdiv(EN, 256), 256, 0, stream>>>(ei, E, EN, asr, adt, (unsigned*)mx3);
  edge_expsum<1><<<cdiv(EN, 256), 256, 0, stream>>>(ei, E, EN, asr, adt, (const unsigned*)mx3, sm3);
  edge_aggr2ch<<<cdiv(EN, 256), 256, 0, stream>>>(ei, E, EN, g3, asr, adt, (const unsigned*)mx3,
                                                  sm3, agg3);
  final_logsoftmax<<<cdiv(N, 256), 256, 0, stream>>>(agg3, b3, (float*)d_out, N);
}